// GraphMLPMixer_38577396253437
// MI455X (gfx1250) — compile-verified
//
#include <hip/hip_runtime.h>
#include <hip/hip_bf16.h>

#define N_NODES 100000
#define M_ENT   200000
#define E_EDGES 600000
#define B_G     128
#define P_P     32
#define BP_     (B_G * P_P)
#define DIN_    64
#define H_      256
#define RW_     16
#define L_      4
#define LM_     4
#define T_      64
#define CH_     (4 * H_)
#define DOUT_   10

typedef float v2f __attribute__((ext_vector_type(2)));
typedef float v8f __attribute__((ext_vector_type(8)));

__device__ __forceinline__ float gelu_tanh(float x) {
    float x3 = x * x * x;
    return 0.5f * x * (1.0f + tanhf(0.7978845608028654f * (x + 0.044715f * x3)));
}

// ---------------------------------------------------------------------------
// Generic WMMA f32 GEMM.  Each wave computes a 32x64 C macro-tile as a 2x4
// grid of 16x16 v_wmma_f32_16x16x4_f32 tiles (8 accumulators).  The K loop is
// branch-free: out-of-range rows/cols load from *clamped* in-bounds addresses
// (their products only reach C entries the guarded epilogue never stores).
//   UNIT=true : sAk==1, A fragment pair is one b64 load w/ immediate offsets
//   SBK != 0  : compile-time B row stride -> +SBK as immediate load offset
// Lane layouts per CDNA5 ISA 7.12.2 (h = lane>>4, l = lane&15):
//   A 16x4 f32 : row l; VGPR0=K(2h), VGPR1=K(2h+1)
//   B 4x16 f32 : col l; VGPR0=K(2h), VGPR1=K(2h+1)
//   C 16x16    : VGPR r -> (row r + 8h, col l)
// ---------------------------------------------------------------------------
#define WMMA_F32(ACC, A_, B_) \
    ACC = __builtin_amdgcn_wmma_f32_16x16x4_f32(false, A_, false, B_, (short)0, ACC, false, false)

template<bool UNIT, int SBK>
__global__ __launch_bounds__(256) void gemm_wmma_kernel(
    const float* __restrict__ A, const float* __restrict__ Bm,
    const float* __restrict__ bias, float* __restrict__ C,
    const int* __restrict__ rowmap,
    int M, int N, int K, int tilesM2, int groupsN, int nBatch,
    long sAm, long sAk, long sAb,
    long sBk, long sBn, long sBb,
    long sCm, long sCn, long sCb,
    int act, int accum)
{
    int wave = (int)((blockIdx.x * (long)blockDim.x + threadIdx.x) >> 5);
    int lane = threadIdx.x & 31;
    int per = tilesM2 * groupsN;
    int batch = wave / per;
    if (batch >= nBatch) return;
    int t  = wave - batch * per;
    int tm = t / groupsN;
    int g  = t - tm * groupsN;

    int half = lane >> 4;
    int l    = lane & 15;

    // two row tiles: row0 .. row0+15, row0+16 .. row0+31
    int row0 = tm * 32;
    int am0 = row0 + l;
    int am1 = row0 + 16 + l;
    int am0c = am0 < M ? am0 : (M - 1);
    int am1c = am1 < M ? am1 : (M - 1);
    long ar0 = rowmap ? (long)rowmap[am0c] : (long)am0c;
    long ar1 = rowmap ? (long)rowmap[am1c] : (long)am1c;

    const long sbk = SBK ? (long)SBK : sBk;

    const float* baseA = A + (long)batch * sAb + (long)(half * 2) * sAk;
    const float* pA0 = baseA + ar0 * sAm;
    const float* pA1 = baseA + ar1 * sAm;
    const long stepA = 4 * sAk;
    const long stepB = 4 * sbk;

    // four col tiles
    const float* baseB = Bm + (long)batch * sBb + (long)(half * 2) * sbk;
    int bn0 = (g * 4 + 0) * 16 + l;
    int bn1 = (g * 4 + 1) * 16 + l;
    int bn2 = (g * 4 + 2) * 16 + l;
    int bn3 = (g * 4 + 3) * 16 + l;
    const float* pB0 = baseB + (long)(bn0 < N ? bn0 : N - 1) * sBn;
    const float* pB1 = baseB + (long)(bn1 < N ? bn1 : N - 1) * sBn;
    const float* pB2 = baseB + (long)(bn2 < N ? bn2 : N - 1) * sBn;
    const float* pB3 = baseB + (long)(bn3 < N ? bn3 : N - 1) * sBn;

    v8f acc00 = {}, acc01 = {}, acc02 = {}, acc03 = {};
    v8f acc10 = {}, acc11 = {}, acc12 = {}, acc13 = {};

    for (int k = 0; k < K; k += 4) {
        v2f a0, a1, b0, b1, b2, b3;
        if (UNIT) {
            a0 = *(const v2f*)pA0;           // sAk==1: contiguous pair, b64 load
            a1 = *(const v2f*)pA1;
        } else {
            a0.x = pA0[0]; a0.y = pA0[sAk];
            a1.x = pA1[0]; a1.y = pA1[sAk];
        }
        b0.x = pB0[0]; b0.y = pB0[sbk];      // SBK!=0: sbk is an immediate offset
        b1.x = pB1[0]; b1.y = pB1[sbk];
        b2.x = pB2[0]; b2.y = pB2[sbk];
        b3.x = pB3[0]; b3.y = pB3[sbk];
        WMMA_F32(acc00, a0, b0);
        WMMA_F32(acc01, a0, b1);
        WMMA_F32(acc02, a0, b2);
        WMMA_F32(acc03, a0, b3);
        WMMA_F32(acc10, a1, b0);
        WMMA_F32(acc11, a1, b1);
        WMMA_F32(acc12, a1, b2);
        WMMA_F32(acc13, a1, b3);
        if (UNIT) { pA0 += 4;     pA1 += 4;     }
        else      { pA0 += stepA; pA1 += stepA; }
        pB0 += stepB; pB1 += stepB; pB2 += stepB; pB3 += stepB;
    }

    float* Cb = C + (long)batch * sCb;
    v8f accs[2][4] = {{acc00, acc01, acc02, acc03}, {acc10, acc11, acc12, acc13}};
    int bns[4] = {bn0, bn1, bn2, bn3};
    #pragma unroll
    for (int j = 0; j < 4; ++j) {
        int n = bns[j];
        if (n >= N) continue;
        float bv = bias ? bias[n] : 0.0f;
        #pragma unroll
        for (int i = 0; i < 2; ++i) {
            int rbase = row0 + i * 16 + half * 8;
            #pragma unroll
            for (int r = 0; r < 8; ++r) {
                int cm = rbase + r;
                if (cm < M) {
                    float v = accs[i][j][r] + bv;
                    if (act == 1)      v = fmaxf(v, 0.0f);
                    else if (act == 2) v = gelu_tanh(v);
                    long off = (long)cm * sCm + (long)n * sCn;
                    if (accum) Cb[off] += v;
                    else       Cb[off]  = v;
                }
            }
        }
    }
}

// ---------------------------------------------------------------------------
// Segment / graph helper kernels
// ---------------------------------------------------------------------------
__global__ void count_kernel(const int* __restrict__ sb, const int* __restrict__ mp,
                             float* cntS, float* cntN, int n) {
    int t = blockIdx.x * blockDim.x + threadIdx.x;
    if (t >= n) return;
    atomicAdd(&cntS[sb[t]], 1.0f);
    atomicAdd(&cntN[mp[t]], 1.0f);
}

__global__ void deg_kernel(const int* __restrict__ dst, float* deg, int n) {
    int t = blockIdx.x * blockDim.x + threadIdx.x;
    if (t >= n) return;
    atomicAdd(&deg[dst[t]], 1.0f);
}

__global__ void nrm_kernel(const float* __restrict__ deg, float* nrm, int n) {
    int t = blockIdx.x * blockDim.x + threadIdx.x;
    if (t >= n) return;
    nrm[t] = rsqrtf(deg[t] + 1.0f);
}

// rows of width H_=256; thread handles 4 consecutive floats of one row
__global__ void scatter_add_rows(const float* __restrict__ v, const int* __restrict__ idx,
                                 float* pool, int R) {
    long t = blockIdx.x * (long)blockDim.x + threadIdx.x;
    long total = (long)R * 64;
    if (t >= total) return;
    int m = (int)(t >> 6);
    int j = ((int)t & 63) * 4;
    int s = idx[m];
    const float4 x = *(const float4*)(v + (long)m * H_ + j);
    float* p = pool + (long)s * H_ + j;
    atomicAdd(p + 0, x.x); atomicAdd(p + 1, x.y);
    atomicAdd(p + 2, x.z); atomicAdd(p + 3, x.w);
}

__global__ void div_rows_kernel(float* pool, const float* __restrict__ cnt, long R) {
    long t = blockIdx.x * (long)blockDim.x + threadIdx.x;
    long total = R * H_;
    if (t >= total) return;
    int r = (int)(t >> 8);
    pool[t] /= fmaxf(cnt[r], 1.0f);
}

__global__ void add_rows_gather(float* h, const float* __restrict__ u,
                                const int* __restrict__ idx, int R) {
    long t = blockIdx.x * (long)blockDim.x + threadIdx.x;
    long total = (long)R * H_;
    if (t >= total) return;
    int m = (int)(t >> 8);
    int j = (int)t & 255;
    h[t] += u[(long)idx[m] * H_ + j];
}

__global__ void gather_rows(const float* __restrict__ pool, const int* __restrict__ idx,
                            float* h, int R) {
    long t = blockIdx.x * (long)blockDim.x + threadIdx.x;
    long total = (long)R * H_;
    if (t >= total) return;
    int m = (int)(t >> 8);
    int j = (int)t & 255;
    h[t] = pool[(long)idx[m] * H_ + j];
}

__global__ void add_kernel(const float* __restrict__ a, const float* __restrict__ b,
                           float* c, long n) {
    long t = blockIdx.x * (long)blockDim.x + threadIdx.x;
    if (t >= n) return;
    c[t] = a[t] + b[t];
}

// agg[dst] += hw[src] * nrm[src]*nrm[dst]; thread handles 4 floats of one edge row
__global__ void edge_scatter_kernel(const float* __restrict__ hw,
                                    const int* __restrict__ src, const int* __restrict__ dst,
                                    const float* __restrict__ nrm, float* agg, int E) {
    long t = blockIdx.x * (long)blockDim.x + threadIdx.x;
    long total = (long)E * 64;
    if (t >= total) return;
    int e = (int)(t >> 6);
    int j = ((int)t & 63) * 4;
    int s = src[e], d = dst[e];
    float w = nrm[s] * nrm[d];
    const float4 v = *(const float4*)(hw + (long)s * H_ + j);
    float* p = agg + (long)d * H_ + j;
    atomicAdd(p + 0, v.x * w); atomicAdd(p + 1, v.y * w);
    atomicAdd(p + 2, v.z * w); atomicAdd(p + 3, v.w * w);
}

__global__ void gcn_finish_kernel(const float* __restrict__ agg, const float* __restrict__ hw,
                                  const float* __restrict__ nrm, const float* __restrict__ bg,
                                  float* h, int R) {
    long t = blockIdx.x * (long)blockDim.x + threadIdx.x;
    long total = (long)R * H_;
    if (t >= total) return;
    int m = (int)(t >> 8);
    int j = (int)t & 255;
    float sn = nrm[m]; sn *= sn;
    h[t] = fmaxf(agg[t] + hw[t] * sn + bg[j], 0.0f);
}

// LayerNorm over last dim H_=256; one wave32 per row
__global__ void ln_kernel(const float* __restrict__ x, float* y,
                          const float* __restrict__ s, const float* __restrict__ b, int R) {
    int wave = (int)((blockIdx.x * (long)blockDim.x + threadIdx.x) >> 5);
    int lane = threadIdx.x & 31;
    if (wave >= R) return;
    const float* xr = x + (long)wave * H_;
    float vals[8];
    float sum = 0.0f;
    #pragma unroll
    for (int i = 0; i < 8; ++i) { vals[i] = xr[lane + 32 * i]; sum += vals[i]; }
    #pragma unroll
    for (int o = 16; o > 0; o >>= 1) sum += __shfl_xor(sum, o, 32);
    float mean = sum * (1.0f / H_);
    float var = 0.0f;
    #pragma unroll
    for (int i = 0; i < 8; ++i) { float d = vals[i] - mean; var += d * d; }
    #pragma unroll
    for (int o = 16; o > 0; o >>= 1) var += __shfl_xor(var, o, 32);
    var *= (1.0f / H_);
    float rs = rsqrtf(var + 1e-5f);
    float* yr = y + (long)wave * H_;
    #pragma unroll
    for (int i = 0; i < 8; ++i) {
        int j = lane + 32 * i;
        yr[j] = (vals[i] - mean) * rs * s[j] + b[j];
    }
}

// masked mean over patches: g[b][h] = sum_p z[b,p,h]*m / sum_p m
__global__ void mask_pool_kernel(const float* __restrict__ z,
                                 const unsigned char* __restrict__ mask, float* g) {
    int t = blockIdx.x * blockDim.x + threadIdx.x;
    if (t >= B_G * H_) return;
    int b = t >> 8;
    int h = t & 255;
    float s = 0.0f, c = 0.0f;
    for (int p = 0; p < P_P; ++p) {
        float mm = mask[b * P_P + p] ? 1.0f : 0.0f;
        s += z[((long)(b * P_P + p)) * H_ + h] * mm;
        c += mm;
    }
    g[t] = s / fmaxf(c, 1.0f);
}

// ---------------------------------------------------------------------------
static void launch_gemm(hipStream_t stream,
                        const float* A, const float* Bm, const float* bias, float* C,
                        const int* rowmap, int M, int N, int K, int nBatch,
                        long sAm, long sAk, long sAb,
                        long sBk, long sBn, long sBb,
                        long sCm, long sCn, long sCb,
                        int act, int accum) {
    int tilesM2 = (M + 31) / 32;            // 32 rows per wave
    int groupsN = ((N + 15) / 16 + 3) / 4;  // 64 cols per wave
    long waves = (long)nBatch * tilesM2 * groupsN;
    long blocks = (waves + 7) / 8;          // 8 waves (256 threads) per block
    dim3 grid((unsigned)blocks);
    #define GEMM_ARGS A, Bm, bias, C, rowmap, M, N, K, tilesM2, groupsN, nBatch, \
                      sAm, sAk, sAb, sBk, sBn, sBb, sCm, sCn, sCb, act, accum
    if (sAk == 1) {
        if (sBk == 256)
            gemm_wmma_kernel<true, 256><<<grid, 256, 0, stream>>>(GEMM_ARGS);
        else if (sBk == 1024)
            gemm_wmma_kernel<true, 1024><<<grid, 256, 0, stream>>>(GEMM_ARGS);
        else
            gemm_wmma_kernel<true, 0><<<grid, 256, 0, stream>>>(GEMM_ARGS);
    } else {
        gemm_wmma_kernel<false, 0><<<grid, 256, 0, stream>>>(GEMM_ARGS);
    }
    #undef GEMM_ARGS
}

static inline unsigned gblk(long n, int bs) { return (unsigned)((n + bs - 1) / bs); }

extern "C" void kernel_launch(void* const* d_in, const int* in_sizes, int n_in,
                              void* d_out, int out_size, void* d_ws, size_t ws_size,
                              hipStream_t stream) {
    const float* x          = (const float*)d_in[0];
    const float* patch_pe   = (const float*)d_in[1];
    const unsigned char* mask = (const unsigned char*)d_in[2];
    const int* nodes_mapper = (const int*)d_in[3];
    const int* sub_batch    = (const int*)d_in[4];
    const int* edge_index   = (const int*)d_in[5];
    const int* e_src = edge_index;
    const int* e_dst = edge_index + E_EDGES;
    const float* W_pre = (const float*)d_in[6];  const float* b_pre = (const float*)d_in[7];
    const float* W_u   = (const float*)d_in[8];  const float* b_u   = (const float*)d_in[9];
    const float* W_g   = (const float*)d_in[10]; const float* b_g   = (const float*)d_in[11];
    const float* Wp1   = (const float*)d_in[12]; const float* bp1   = (const float*)d_in[13];
    const float* Wp2   = (const float*)d_in[14]; const float* bp2   = (const float*)d_in[15];
    const float* ln1_s = (const float*)d_in[16]; const float* ln1_b = (const float*)d_in[17];
    const float* Wt1   = (const float*)d_in[18]; const float* bt1   = (const float*)d_in[19];
    const float* Wt2   = (const float*)d_in[20]; const float* bt2   = (const float*)d_in[21];
    const float* ln2_s = (const float*)d_in[22]; const float* ln2_b = (const float*)d_in[23];
    const float* Wc1   = (const float*)d_in[24]; const float* bc1   = (const float*)d_in[25];
    const float* Wc2   = (const float*)d_in[26]; const float* bc2   = (const float*)d_in[27];
    const float* ln_f_s = (const float*)d_in[28]; const float* ln_f_b = (const float*)d_in[29];
    const float* W_head = (const float*)d_in[30]; const float* b_head = (const float*)d_in[31];
    float* out = (float*)d_out;
    (void)in_sizes; (void)n_in; (void)out_size; (void)ws_size;

    // ---- workspace carve-up -------------------------------------------------
    char* wp = (char*)d_ws;
    auto alloc = [&](size_t nfloats) -> float* {
        float* p = (float*)wp;
        size_t bytes = (nfloats * sizeof(float) + 255) & ~(size_t)255;
        wp += bytes;
        return p;
    };
    float* hN    = alloc((size_t)N_NODES * H_);
    float* h     = alloc((size_t)M_ENT * H_);
    float* hw    = alloc((size_t)M_ENT * H_);
    float* agg   = alloc((size_t)M_ENT * H_);
    float* poolN = agg;  // alias: poolN fully consumed before agg is re-zeroed
    float* subp  = alloc((size_t)BP_ * H_);
    float* bufA  = alloc((size_t)BP_ * H_);
    float* bufB  = alloc((size_t)BP_ * H_);
    float* z     = alloc((size_t)BP_ * H_);
    float* zl    = alloc((size_t)BP_ * H_);
    float* t1    = alloc((size_t)B_G * H_ * T_);
    float* c1    = alloc((size_t)BP_ * CH_);
    float* deg   = alloc(M_ENT);
    float* nrm   = alloc(M_ENT);
    float* cntS  = alloc(BP_);
    float* cntN  = alloc(N_NODES);
    float* gbuf  = alloc((size_t)B_G * H_);
    float* gln   = alloc((size_t)B_G * H_);

    // ---- segment counts & GCN degree normalization --------------------------
    hipMemsetAsync(deg, 0, M_ENT * sizeof(float), stream);
    hipMemsetAsync(cntS, 0, BP_ * sizeof(float), stream);
    hipMemsetAsync(cntN, 0, N_NODES * sizeof(float), stream);
    count_kernel<<<gblk(M_ENT, 256), 256, 0, stream>>>(sub_batch, nodes_mapper, cntS, cntN, M_ENT);
    deg_kernel<<<gblk(E_EDGES, 256), 256, 0, stream>>>(e_dst, deg, E_EDGES);
    nrm_kernel<<<gblk(M_ENT, 256), 256, 0, stream>>>(deg, nrm, M_ENT);

    // ---- pre_mp: hN = relu(x @ W_pre + b_pre) -------------------------------
    launch_gemm(stream, x, W_pre, b_pre, hN, nullptr, N_NODES, H_, DIN_, 1,
                DIN_, 1, 0,  H_, 1, 0,  H_, 1, 0,  1, 0);

    // ---- GCN layers ---------------------------------------------------------
    for (int i = 0; i < L_; ++i) {
        if (i > 0) {
            // pre_layer_scatter: pool -> U-MLP on pooled (gather commutes) -> add
            hipMemsetAsync(subp, 0, (size_t)BP_ * H_ * sizeof(float), stream);
            scatter_add_rows<<<gblk((long)M_ENT * 64, 256), 256, 0, stream>>>(h, sub_batch, subp, M_ENT);
            div_rows_kernel<<<gblk((long)BP_ * H_, 256), 256, 0, stream>>>(subp, cntS, BP_);
            launch_gemm(stream, subp, W_u + (size_t)(i - 1) * H_ * H_, b_u + (i - 1) * H_,
                        bufB, nullptr, BP_, H_, H_, 1,
                        H_, 1, 0,  H_, 1, 0,  H_, 1, 0,  1, 0);
            add_rows_gather<<<gblk((long)M_ENT * H_, 256), 256, 0, stream>>>(h, bufB, sub_batch, M_ENT);
            // post_layer_scatter: mean over duplicated nodes, broadcast back
            hipMemsetAsync(poolN, 0, (size_t)N_NODES * H_ * sizeof(float), stream);
            scatter_add_rows<<<gblk((long)M_ENT * 64, 256), 256, 0, stream>>>(h, nodes_mapper, poolN, M_ENT);
            div_rows_kernel<<<gblk((long)N_NODES * H_, 256), 256, 0, stream>>>(poolN, cntN, N_NODES);
            gather_rows<<<gblk((long)M_ENT * H_, 256), 256, 0, stream>>>(poolN, nodes_mapper, h, M_ENT);
        }
        // hw = h @ W_g[i]   (layer 0: gather from hN fused via rowmap)
        const float* Ain = (i == 0) ? hN : h;
        const int* rmap = (i == 0) ? nodes_mapper : nullptr;
        launch_gemm(stream, Ain, W_g + (size_t)i * H_ * H_, nullptr, hw, rmap,
                    M_ENT, H_, H_, 1,
                    H_, 1, 0,  H_, 1, 0,  H_, 1, 0,  0, 0);
        hipMemsetAsync(agg, 0, (size_t)M_ENT * H_ * sizeof(float), stream);
        edge_scatter_kernel<<<gblk((long)E_EDGES * 64, 256), 256, 0, stream>>>(hw, e_src, e_dst, nrm, agg, E_EDGES);
        gcn_finish_kernel<<<gblk((long)M_ENT * H_, 256), 256, 0, stream>>>(agg, hw, nrm, b_g + i * H_, h, M_ENT);
    }

    // ---- final scatter to [BP, H] -------------------------------------------
    hipMemsetAsync(subp, 0, (size_t)BP_ * H_ * sizeof(float), stream);
    scatter_add_rows<<<gblk((long)M_ENT * 64, 256), 256, 0, stream>>>(h, sub_batch, subp, M_ENT);
    div_rows_kernel<<<gblk((long)BP_ * H_, 256), 256, 0, stream>>>(subp, cntS, BP_);

    // ---- patch RW-PE 2-layer MLP, then z = pooled + pe ----------------------
    launch_gemm(stream, patch_pe, Wp1, bp1, bufA, nullptr, BP_, H_, RW_, 1,
                RW_, 1, 0,  H_, 1, 0,  H_, 1, 0,  1, 0);
    launch_gemm(stream, bufA, Wp2, bp2, bufB, nullptr, BP_, H_, H_, 1,
                H_, 1, 0,  H_, 1, 0,  H_, 1, 0,  1, 0);
    add_kernel<<<gblk((long)BP_ * H_, 256), 256, 0, stream>>>(subp, bufB, z, (long)BP_ * H_);

    // ---- MLP-Mixer ----------------------------------------------------------
    for (int l = 0; l < LM_; ++l) {
        // token mixing: LN, then per-batch [H,P]@[P,T] (A read transposed), gelu,
        // then [H,T]@[T,P] accumulated back into z with transposed C strides.
        ln_kernel<<<gblk((long)BP_ * 32, 256), 256, 0, stream>>>(z, zl, ln1_s + l * H_, ln1_b + l * H_, BP_);
        launch_gemm(stream, zl, Wt1 + (size_t)l * P_P * T_, bt1 + l * T_, t1, nullptr,
                    H_, T_, P_P, B_G,
                    /*A*/ 1, H_, (long)P_P * H_,
                    /*B*/ T_, 1, 0,
                    /*C*/ T_, 1, (long)H_ * T_,  2, 0);
        launch_gemm(stream, t1, Wt2 + (size_t)l * T_ * P_P, bt2 + l * P_P, z, nullptr,
                    H_, P_P, T_, B_G,
                    /*A*/ T_, 1, (long)H_ * T_,
                    /*B*/ P_P, 1, 0,
                    /*C*/ 1, H_, (long)P_P * H_,  0, 1);
        // channel mixing
        ln_kernel<<<gblk((long)BP_ * 32, 256), 256, 0, stream>>>(z, zl, ln2_s + l * H_, ln2_b + l * H_, BP_);
        launch_gemm(stream, zl, Wc1 + (size_t)l * H_ * CH_, bc1 + l * CH_, c1, nullptr,
                    BP_, CH_, H_, 1,
                    H_, 1, 0,  CH_, 1, 0,  CH_, 1, 0,  2, 0);
        launch_gemm(stream, c1, Wc2 + (size_t)l * CH_ * H_, bc2 + l * H_, z, nullptr,
                    BP_, H_, CH_, 1,
                    CH_, 1, 0,  H_, 1, 0,  H_, 1, 0,  0, 1);
    }

    // ---- masked mean pool, final LN, head -----------------------------------
    mask_pool_kernel<<<gblk(B_G * H_, 256), 256, 0, stream>>>(z, mask, gbuf);
    ln_kernel<<<gblk((long)B_G * 32, 256), 256, 0, stream>>>(gbuf, gln, ln_f_s, ln_f_b, B_G);
    launch_gemm(stream, gln, W_head, b_head, out, nullptr, B_G, DOUT_, H_, 1,
                H_, 1, 0,  DOUT_, 1, 0,  DOUT_, 1, 0,  0, 0);
}